// AllSetTransformer_64072322122411
// MI455X (gfx1250) — compile-verified
//
#include <hip/hip_runtime.h>
#include <math.h>

#define NN 100000   // nodes
#define NE 30000    // hyperedges
#define NP 400000   // pins
#define DD 128      // hidden
#define NH 4        // heads
#define HK 32       // per-head dim
#define PROJ_W 144  // 128 V-cols + 4 logit cols + 12 zero pad (9 x 16 tiles)
#define LNP (DD + 1) // LDS row pad to avoid bank conflicts

typedef float v2f __attribute__((ext_vector_type(2)));
typedef float v8f __attribute__((ext_vector_type(8)));

__device__ __forceinline__ v8f wmma_f32_k4(v2f a, v2f b, v8f c) {
  // V_WMMA_F32_16X16X4_F32 : D = A(16x4) * B(4x16) + C(16x16), all f32
  return __builtin_amdgcn_wmma_f32_16x16x4_f32(false, a, false, b, (short)0, c,
                                               false, false);
}

// ---- monotonic float <-> uint encoding for atomic max over floats ----
__device__ __forceinline__ unsigned fenc(float f) {
  unsigned u = __float_as_uint(f);
  return (u & 0x80000000u) ? ~u : (u | 0x80000000u);
}
__device__ __forceinline__ float fdec(unsigned u) {
  return __uint_as_float((u & 0x80000000u) ? (u & 0x7FFFFFFFu) : ~u);
}

// ---------------------------------------------------------------------------
// Pack B = [ Vw_flat (128x128) | Kw@Qw (128x4) | zeros (128x12) ] per block.
// Vw_flat[d][h*32+k] = Vw[b][h][d][k]; qhat[d][h] = sum_k Kw[b][h][d][k]*Qw[b][h][k]
// ---------------------------------------------------------------------------
__global__ void repack_kernel(const float* __restrict__ Kw,
                              const float* __restrict__ Vw,
                              const float* __restrict__ Qw,
                              float* __restrict__ Bp) {
  const int total = 4 * DD * PROJ_W;
  for (int t = blockIdx.x * blockDim.x + threadIdx.x; t < total;
       t += gridDim.x * blockDim.x) {
    int b = t / (DD * PROJ_W);
    int r = t - b * DD * PROJ_W;
    int d = r / PROJ_W;
    int j = r - d * PROJ_W;
    float val = 0.0f;
    if (j < DD) {
      int h = j >> 5, k = j & 31;
      val = Vw[(((size_t)b * NH + h) * DD + d) * HK + k];
    } else if (j < DD + NH) {
      int h = j - DD;
      const float* kp = Kw + (((size_t)b * NH + h) * DD + d) * HK;
      const float* qp = Qw + ((size_t)b * NH + h) * HK;
      float s = 0.0f;
#pragma unroll
      for (int k = 0; k < HK; ++k) s += kp[k] * qp[k];
      val = s;
    }
    Bp[t] = val;
  }
}

// ---------------------------------------------------------------------------
// Reset per-block accumulators: agg=0, max-encoding=0 (== -inf), denom=0
// ---------------------------------------------------------------------------
__global__ void fill_kernel(float* __restrict__ agg, unsigned* __restrict__ menc,
                            float* __restrict__ denom, int n_tgt) {
  int t = blockIdx.x * blockDim.x + threadIdx.x;
  if (t < n_tgt * DD) agg[t] = 0.0f;
  if (t < n_tgt * NH) { menc[t] = 0u; denom[t] = 0.0f; }
}

// ---------------------------------------------------------------------------
// C[M,144] = A[M,128] @ B[128,144]    (one wave per 16x16 tile, f32 WMMA k=4)
// ---------------------------------------------------------------------------
__global__ __launch_bounds__(256) void gemm144_kernel(
    const float* __restrict__ A, const float* __restrict__ B,
    float* __restrict__ C, int mtiles) {
  int gw = (blockIdx.x * blockDim.x + threadIdx.x) >> 5;
  if (gw >= mtiles * 9) return;
  int lane = threadIdx.x & 31;
  int mt = gw / 9, nt = gw - mt * 9;
  int row0 = mt * 16, col0 = nt * 16;
  int m = lane & 15, hi = lane >> 4;
  const float* Ar = A + (size_t)(row0 + m) * DD;
  v8f c;
#pragma unroll
  for (int i = 0; i < 8; ++i) c[i] = 0.0f;
#pragma unroll 4
  for (int kb = 0; kb < DD; kb += 4) {
    int k0 = kb + (hi ? 2 : 0);
    v2f a; a[0] = Ar[k0]; a[1] = Ar[k0 + 1];
    v2f bb;
    bb[0] = B[(size_t)k0 * PROJ_W + col0 + m];
    bb[1] = B[(size_t)(k0 + 1) * PROJ_W + col0 + m];
    c = wmma_f32_k4(a, bb, c);
  }
#pragma unroll
  for (int j = 0; j < 8; ++j)
    C[(size_t)(row0 + j + hi * 8) * PROJ_W + col0 + m] = c[j];
}

// ---------------------------------------------------------------------------
// Pin pass 1: segment max of logits (encoded-uint atomic max)
// ---------------------------------------------------------------------------
__global__ void pin_max_kernel(const int* __restrict__ src,
                               const int* __restrict__ tgt,
                               const float* __restrict__ proj,
                               unsigned* __restrict__ menc) {
  int t = blockIdx.x * blockDim.x + threadIdx.x;
  if (t >= NP * NH) return;
  int pin = t >> 2, h = t & 3;
  float l = proj[(size_t)src[pin] * PROJ_W + DD + h];
  atomicMax(&menc[(size_t)tgt[pin] * NH + h], fenc(l));
}

// ---------------------------------------------------------------------------
// Pin pass 2: ex = exp(l - max); denom += ex; cache ex per pin
// ---------------------------------------------------------------------------
__global__ void pin_exp_kernel(const int* __restrict__ src,
                               const int* __restrict__ tgt,
                               const float* __restrict__ proj,
                               const unsigned* __restrict__ menc,
                               float* __restrict__ denom,
                               float* __restrict__ pinw) {
  int t = blockIdx.x * blockDim.x + threadIdx.x;
  if (t >= NP * NH) return;
  int pin = t >> 2, h = t & 3;
  int tg = tgt[pin];
  float l = proj[(size_t)src[pin] * PROJ_W + DD + h];
  float e = __expf(l - fdec(menc[(size_t)tg * NH + h]));
  pinw[t] = e;
  atomicAdd(&denom[(size_t)tg * NH + h], e);
}

// ---------------------------------------------------------------------------
// Pin pass 3: agg[tgt] += (ex/denom) * Vx[src]   (32 lanes per pin, float4)
// ---------------------------------------------------------------------------
__global__ void pin_scatter_kernel(const int* __restrict__ src,
                                   const int* __restrict__ tgt,
                                   const float* __restrict__ proj,
                                   const float* __restrict__ denom,
                                   const float* __restrict__ pinw,
                                   float* __restrict__ agg) {
  int gt = blockIdx.x * blockDim.x + threadIdx.x;
  int pin = gt >> 5;
  if (pin >= NP) return;
  int lane = gt & 31;
  int s = src[pin], tg = tgt[pin];
  int h = lane >> 3;  // channels lane*4..lane*4+3 all belong to head lane>>3
  float w = pinw[(size_t)pin * NH + h] / denom[(size_t)tg * NH + h];
  const float4 v = *(const float4*)(proj + (size_t)s * PROJ_W + lane * 4);
  float* ap = agg + (size_t)tg * DD + lane * 4;
  atomicAdd(ap + 0, w * v.x);
  atomicAdd(ap + 1, w * v.y);
  atomicAdd(ap + 2, w * v.z);
  atomicAdd(ap + 3, w * v.w);
}

// ---------------------------------------------------------------------------
// LayerNorm over 16 rows held in LDS; 8 threads per row, 16 elems per thread
// ---------------------------------------------------------------------------
__device__ __forceinline__ void layernorm_tile(float X[16][LNP],
                                               const float* __restrict__ g,
                                               const float* __restrict__ bta,
                                               float rs[16][8], float rq[16][8],
                                               int tid) {
  int row = tid >> 3, base = (tid & 7) * 16, s = tid & 7;
  float sm = 0.0f, sq = 0.0f;
#pragma unroll
  for (int i = 0; i < 16; ++i) {
    float v = X[row][base + i];
    sm += v; sq += v * v;
  }
  rs[row][s] = sm; rq[row][s] = sq;
  __syncthreads();
  sm = 0.0f; sq = 0.0f;
#pragma unroll
  for (int j = 0; j < 8; ++j) { sm += rs[row][j]; sq += rq[row][j]; }
  float mu = sm * (1.0f / DD);
  float var = sq * (1.0f / DD) - mu * mu;
  float rstd = rsqrtf(var + 1e-5f);
#pragma unroll
  for (int i = 0; i < 16; ++i)
    X[row][base + i] = (X[row][base + i] - mu) * rstd * g[base + i] + bta[base + i];
  __syncthreads();
}

// ---------------------------------------------------------------------------
// Post: x = LN0(agg + Q); h = relu(x@W1+b1)@W2+b2; out = relu(LN1(x+h))
// One 128-thread block per 16 target rows; GEMMs via f32 WMMA with A in LDS.
// ---------------------------------------------------------------------------
__global__ __launch_bounds__(128) void post_kernel(
    const float* __restrict__ agg, const float* __restrict__ qflat,
    const float* __restrict__ W1, const float* __restrict__ b1,
    const float* __restrict__ W2, const float* __restrict__ b2,
    const float* __restrict__ g0, const float* __restrict__ be0,
    const float* __restrict__ g1, const float* __restrict__ be1,
    float* __restrict__ out, int n_tgt) {
  __shared__ float xs[16][LNP];
  __shared__ float hs[16][LNP];
  __shared__ float rs[16][8];
  __shared__ float rq[16][8];
  (void)n_tgt;

  int tid = threadIdx.x;
  int row0 = blockIdx.x * 16;
  int row = tid >> 3, base = (tid & 7) * 16;

  // load agg + learned-query skip connection
#pragma unroll
  for (int i = 0; i < 16; ++i)
    xs[row][base + i] =
        agg[(size_t)(row0 + row) * DD + base + i] + qflat[base + i];
  __syncthreads();
  layernorm_tile(xs, g0, be0, rs, rq, tid);  // xs = LN0

  int lane = tid & 31, wave = tid >> 5;
  int m = lane & 15, hi = lane >> 4;

  // GEMM1: hs = relu(xs @ W1 + b1); each wave owns 2 column tiles
#pragma unroll
  for (int p = 0; p < 2; ++p) {
    int col0 = (wave + p * 4) * 16;
    v8f c;
#pragma unroll
    for (int i = 0; i < 8; ++i) c[i] = 0.0f;
#pragma unroll 4
    for (int kb = 0; kb < DD; kb += 4) {
      int k0 = kb + (hi ? 2 : 0);
      v2f a; a[0] = xs[m][k0]; a[1] = xs[m][k0 + 1];
      v2f bb;
      bb[0] = W1[(size_t)k0 * DD + col0 + m];
      bb[1] = W1[(size_t)(k0 + 1) * DD + col0 + m];
      c = wmma_f32_k4(a, bb, c);
    }
#pragma unroll
    for (int j = 0; j < 8; ++j)
      hs[j + hi * 8][col0 + m] = fmaxf(c[j] + b1[col0 + m], 0.0f);
  }
  __syncthreads();

  // GEMM2: accumulate h2 = hs @ W2 in registers (hs still being read by all)
  v8f c2[2];
#pragma unroll
  for (int p = 0; p < 2; ++p) {
    int col0 = (wave + p * 4) * 16;
#pragma unroll
    for (int i = 0; i < 8; ++i) c2[p][i] = 0.0f;
#pragma unroll 4
    for (int kb = 0; kb < DD; kb += 4) {
      int k0 = kb + (hi ? 2 : 0);
      v2f a; a[0] = hs[m][k0]; a[1] = hs[m][k0 + 1];
      v2f bb;
      bb[0] = W2[(size_t)k0 * DD + col0 + m];
      bb[1] = W2[(size_t)(k0 + 1) * DD + col0 + m];
      c2[p] = wmma_f32_k4(a, bb, c2[p]);
    }
  }
  __syncthreads();  // all reads of hs complete before overwrite

  // residual: hs = xs + h2 + b2
#pragma unroll
  for (int p = 0; p < 2; ++p) {
    int col0 = (wave + p * 4) * 16;
#pragma unroll
    for (int j = 0; j < 8; ++j) {
      int r = j + hi * 8, cc = col0 + m;
      hs[r][cc] = xs[r][cc] + c2[p][j] + b2[cc];
    }
  }
  __syncthreads();
  layernorm_tile(hs, g1, be1, rs, rq, tid);  // hs = LN1

  // relu + store block output
#pragma unroll
  for (int i = 0; i < 16; ++i)
    out[(size_t)(row0 + row) * DD + base + i] = fmaxf(hs[row][base + i], 0.0f);
}

// ---------------------------------------------------------------------------
extern "C" void kernel_launch(void* const* d_in, const int* in_sizes, int n_in,
                              void* d_out, int out_size, void* d_ws,
                              size_t ws_size, hipStream_t stream) {
  (void)in_sizes; (void)n_in; (void)out_size; (void)ws_size;

  const float* x0in = (const float*)d_in[0];
  const float* Kw   = (const float*)d_in[1];
  const float* Vw   = (const float*)d_in[2];
  const float* Qw   = (const float*)d_in[3];
  const float* W1   = (const float*)d_in[4];
  const float* b1   = (const float*)d_in[5];
  const float* W2   = (const float*)d_in[6];
  const float* b2   = (const float*)d_in[7];
  const float* g0   = (const float*)d_in[8];
  const float* be0  = (const float*)d_in[9];
  const float* g1   = (const float*)d_in[10];
  const float* be1  = (const float*)d_in[11];
  const int* node_idx  = (const int*)d_in[12];
  const int* hedge_idx = (const int*)d_in[13];

  // d_out = [x0 (nodes) | x1 (hyperedges)]; block outputs land here directly
  float* xNode = (float*)d_out;
  float* xEdge = xNode + (size_t)NN * DD;

  // workspace partition (floats)
  float* ws = (float*)d_ws;
  size_t o = 0;
  float* proj = ws + o;                 o += (size_t)NN * PROJ_W;  // Vx|logits
  float* agg  = ws + o;                 o += (size_t)NN * DD;
  unsigned* menc = (unsigned*)(ws + o); o += (size_t)NN * NH;
  float* denom = ws + o;                o += (size_t)NN * NH;
  float* pinw  = ws + o;                o += (size_t)NP * NH;
  float* Bp    = ws + o;                o += (size_t)4 * DD * PROJ_W;

  repack_kernel<<<288, 256, 0, stream>>>(Kw, Vw, Qw, Bp);

  for (int b = 0; b < 4; ++b) {
    const bool v2e = (b & 1) == 0;  // vertex->hyperedge
    const float* src_feat = (b == 0) ? x0in : (v2e ? xNode : xEdge);
    float* tgt_feat = v2e ? xEdge : xNode;
    const int* sidx = v2e ? node_idx : hedge_idx;
    const int* tidx = v2e ? hedge_idx : node_idx;
    const int n_src = v2e ? NN : NE;
    const int n_tgt = v2e ? NE : NN;

    const int mtiles = n_src / 16;  // 6250 / 1875, both exact
    const int waves = mtiles * 9;
    gemm144_kernel<<<(waves * 32 + 255) / 256, 256, 0, stream>>>(
        src_feat, Bp + (size_t)b * DD * PROJ_W, proj, mtiles);

    fill_kernel<<<(n_tgt * DD + 255) / 256, 256, 0, stream>>>(agg, menc, denom,
                                                              n_tgt);
    pin_max_kernel<<<(NP * NH + 255) / 256, 256, 0, stream>>>(sidx, tidx, proj,
                                                              menc);
    pin_exp_kernel<<<(NP * NH + 255) / 256, 256, 0, stream>>>(
        sidx, tidx, proj, menc, denom, pinw);
    pin_scatter_kernel<<<(NP * 32 + 255) / 256, 256, 0, stream>>>(
        sidx, tidx, proj, denom, pinw, agg);

    post_kernel<<<n_tgt / 16, 128, 0, stream>>>(
        agg, Qw + (size_t)b * DD, W1 + (size_t)b * DD * DD, b1 + (size_t)b * DD,
        W2 + (size_t)b * DD * DD, b2 + (size_t)b * DD, g0 + (size_t)b * DD,
        be0 + (size_t)b * DD, g1 + (size_t)b * DD, be1 + (size_t)b * DD,
        tgt_feat, n_tgt);
  }
}